// OriginalMultinomial_61933428415670
// MI455X (gfx1250) — compile-verified
//
#include <hip/hip_runtime.h>
#include <stdint.h>

// ---------------------------------------------------------------------------
// Multinomial-without-replacement via Gumbel top-k (B=1024, V=128000, K=64)
// One workgroup (256 threads, 8 wave32) per batch row.
// Pass A: async-DMA row tiles into LDS, build 4096-bin histogram of
//         order-preserved key bits.  Pass B: collect candidates >= threshold
//         bin, then 64x parallel argmax -> sorted top-64 indices.
// ---------------------------------------------------------------------------

#define BATCH      1024
#define VOCAB      128000
#define TOPK       64
#define NBINS      4096
#define BPT        (NBINS / 256)       // bins per thread in scan = 16
#define CAND_MAX   2048
#define TILE       1024                // floats per tile = 256 thr * 4
#define NTILES     (VOCAB / TILE)      // 125 exact

// The async-to-LDS builtin takes v4i pointers in AS(1)/AS(3).
typedef int v4i __attribute__((ext_vector_type(4)));
typedef __attribute__((address_space(1))) v4i* gas1_v4i;
typedef __attribute__((address_space(3))) v4i* gas3_v4i;

__device__ __forceinline__ gas1_v4i to_global(const void* p) {
  return (gas1_v4i)(uintptr_t)p;                  // generic VA == AS1 VA
}
__device__ __forceinline__ gas3_v4i to_lds(void* p) {
  return (gas3_v4i)(uint32_t)(uintptr_t)p;        // low 32b of generic = LDS offset
}

#if defined(__gfx1250__) && __has_builtin(__builtin_amdgcn_global_load_async_to_lds_b128)
#define HAVE_ASYNC_LDS 1
#else
#define HAVE_ASYNC_LDS 0
#endif

#if __has_builtin(__builtin_amdgcn_s_wait_asynccnt)
#define WAIT_ASYNC(n) __builtin_amdgcn_s_wait_asynccnt(n)
#else
#define WAIT_ASYNC(n) asm volatile("s_wait_asynccnt " #n ::: "memory")
#endif

#if __has_builtin(__builtin_amdgcn_s_wait_dscnt)
#define WAIT_DS0() __builtin_amdgcn_s_wait_dscnt(0)
#else
#define WAIT_DS0() asm volatile("s_wait_dscnt 0" ::: "memory")
#endif

// Counter-based RNG: pure function of (row, col) -> identical in both passes.
__device__ __forceinline__ uint32_t pcg_hash(uint32_t x) {
  x = x * 747796405u + 2891336453u;
  uint32_t w = ((x >> ((x >> 28u) + 4u)) ^ x) * 277803737u;
  return (w >> 22u) ^ w;
}

// Order-preserving bits of (log p + Gumbel noise).
__device__ __forceinline__ uint32_t key_bits(float p, uint32_t row, uint32_t col) {
  uint32_t h = pcg_hash(col * 0x9E3779B1u + row * 0x85EBCA77u + 0xC0FFEE37u);
  float u = ((float)(h >> 8) + 0.5f) * (1.0f / 16777216.0f);   // (0,1)
  float g = -__logf(-__logf(u));                               // Gumbel(0,1)
  float key = __logf(p) + g;
  uint32_t b = __float_as_uint(key);
  return (b & 0x80000000u) ? ~b : (b | 0x80000000u);
}

__global__ __launch_bounds__(256)
void gumbel_topk_kernel(const float* __restrict__ probs, int* __restrict__ out) {
  __shared__ uint32_t hist[NBINS];
#if HAVE_ASYNC_LDS
  __shared__ __align__(16) float stage[2][TILE];
#endif
  __shared__ uint32_t chunkScan[256];
  __shared__ uint32_t candKey[CAND_MAX];
  __shared__ uint32_t candIdx[CAND_MAX];
  __shared__ uint32_t s_candCount;
  __shared__ uint32_t s_threshBin;
  __shared__ unsigned long long s_best[8];

  const uint32_t tid = threadIdx.x;
  const uint32_t row = blockIdx.x;
  const float* rowp = probs + (size_t)row * VOCAB;

  for (uint32_t b = tid; b < NBINS; b += 256) hist[b] = 0u;
  if (tid == 0) s_candCount = 0u;
  __syncthreads();

  // ------------------- Pass A: histogram of perturbed keys -----------------
#if HAVE_ASYNC_LDS
  // prime tile 0 (each lane DMAs its own 16B into the staging buffer)
  __builtin_amdgcn_global_load_async_to_lds_b128(
      to_global(rowp + tid * 4), to_lds(&stage[0][tid * 4]), 0, 0);
  for (int t = 0; t < NTILES; ++t) {
    const int buf = t & 1;
    WAIT_ASYNC(0);                                  // tile t staged in LDS
    float4 v = *(const float4*)&stage[buf][tid * 4];
    WAIT_DS0();                                     // v in VGPRs (WAR-safe)
    if (t + 1 < NTILES) {
      __builtin_amdgcn_global_load_async_to_lds_b128(
          to_global(rowp + (size_t)(t + 1) * TILE + tid * 4),
          to_lds(&stage[buf ^ 1][tid * 4]), 0, 0);  // DMA overlaps math below
    }
    const uint32_t col = (uint32_t)t * TILE + tid * 4u;
    const float* vf = (const float*)&v;
#pragma unroll
    for (int j = 0; j < 4; ++j) {
      uint32_t kb = key_bits(vf[j], row, col + j);
      atomicAdd(&hist[kb >> 20], 1u);
    }
  }
#else
  for (int t = 0; t < NTILES; ++t) {
    float4 v = *(const float4*)(rowp + (size_t)t * TILE + tid * 4);
    const uint32_t col = (uint32_t)t * TILE + tid * 4u;
    const float* vf = (const float*)&v;
#pragma unroll
    for (int j = 0; j < 4; ++j) {
      uint32_t kb = key_bits(vf[j], row, col + j);
      atomicAdd(&hist[kb >> 20], 1u);
    }
  }
#endif
  __syncthreads();

  // -------- find threshold bin: suffix counts from the top bin down --------
  {
    uint32_t csum = 0;
#pragma unroll
    for (int j = 0; j < BPT; ++j)
      csum += hist[NBINS - 1 - (tid * BPT + j)];
    chunkScan[tid] = csum;
    __syncthreads();
    for (int off = 1; off < 256; off <<= 1) {       // Hillis-Steele inclusive
      uint32_t mine = chunkScan[tid];
      uint32_t add = (tid >= (uint32_t)off) ? chunkScan[tid - off] : 0u;
      __syncthreads();
      chunkScan[tid] = mine + add;
      __syncthreads();
    }
    uint32_t incl = chunkScan[tid];
    uint32_t excl = (tid == 0) ? 0u : chunkScan[tid - 1];
    if (incl >= TOPK && excl < TOPK) {              // my chunk holds the cut
      uint32_t running = excl, tb = NBINS - 1 - (tid * BPT + BPT - 1);
      for (int j = 0; j < BPT; ++j) {
        uint32_t b = NBINS - 1 - (tid * BPT + j);
        uint32_t c = hist[b];
        if (running + c >= TOPK) { tb = b; break; }
        running += c;
      }
      s_threshBin = tb;
    }
    __syncthreads();
  }
  const uint32_t thresh = s_threshBin << 20;

  // --------- Pass B: collect candidates (row is L2-hot by now) -------------
  for (int t = 0; t < NTILES; ++t) {
    float4 v = *(const float4*)(rowp + (size_t)t * TILE + tid * 4);
    const uint32_t col = (uint32_t)t * TILE + tid * 4u;
    const float* vf = (const float*)&v;
#pragma unroll
    for (int j = 0; j < 4; ++j) {
      uint32_t kb = key_bits(vf[j], row, col + j);
      if (kb >= thresh) {
        uint32_t pos = atomicAdd(&s_candCount, 1u);
        if (pos < CAND_MAX) { candKey[pos] = kb; candIdx[pos] = col + j; }
      }
    }
  }
  __syncthreads();
  const uint32_t ncand = min(s_candCount, (uint32_t)CAND_MAX);

  // --------- 64x parallel argmax -> indices in descending-key order --------
  const uint32_t lane = tid & 31u;
  const uint32_t wv = tid >> 5;
  for (int k = 0; k < TOPK; ++k) {
    unsigned long long best = 0ull;
    for (uint32_t i = tid; i < ncand; i += 256) {
      unsigned long long pk =
          ((unsigned long long)candKey[i] << 32) | (unsigned long long)(i + 1u);
      if (pk > best) best = pk;
    }
#pragma unroll
    for (int off = 16; off > 0; off >>= 1) {        // wave32 shuffle reduce
      unsigned long long o = __shfl_down(best, (unsigned)off, 32);
      if (o > best) best = o;
    }
    if (lane == 0) s_best[wv] = best;
    __syncthreads();
    if (tid == 0) {
      unsigned long long b = 0ull;
      for (int w = 0; w < 8; ++w)
        if (s_best[w] > b) b = s_best[w];
      if (b) {
        uint32_t pos = (uint32_t)(b & 0xFFFFFFFFu) - 1u;
        out[(size_t)row * TOPK + k] = (int)candIdx[pos];
        candKey[pos] = 0u;                          // invalidate winner
      } else {
        out[(size_t)row * TOPK + k] = 0;
      }
    }
    __syncthreads();
  }
}

extern "C" void kernel_launch(void* const* d_in, const int* in_sizes, int n_in,
                              void* d_out, int out_size, void* d_ws, size_t ws_size,
                              hipStream_t stream) {
  (void)in_sizes; (void)n_in; (void)out_size; (void)d_ws; (void)ws_size;
  const float* probs = (const float*)d_in[0];
  int* out = (int*)d_out;   // reference output dtype is int32
  gumbel_topk_kernel<<<dim3(BATCH), dim3(256), 0, stream>>>(probs, out);
}